// DIN_66623532695809
// MI455X (gfx1250) — compile-verified
//
#include <hip/hip_runtime.h>

// MI455X / gfx1250 fused kernel.
// out[b,i,:] = sum_{j<=i} score(b,i,j) * h[b,j,:]
// score(b,i,j) = sum_a lrelu( h_j @ (w1a + h_i*w1c)[:,a] + c_i[a] + b1[a] ) * w2[a] + b2
// with h = x @ ln_w + ln_b, c_i = h_i @ w1b.
//
// Strategy: 1 workgroup = (b, 16-row i-tile), 1 wave32 = one i.
//  - Stage 1: all 16 waves cooperatively build h[b] in LDS via fp32 WMMA
//    (V_WMMA_F32_16X16X4_F32), tiles of x (A-layout) x ln_w (B-layout).
//  - Stage 2: each wave folds h_i into the weights (B_i = w1a + h_i (*) w1c,
//    kept in registers in WMMA-B layout), then GEMMs h-tiles against it over
//    the causal j-range. Epilogue (bias, leaky-relu, w2 dot, causal mask,
//    score @ h accumulation) stays entirely in registers/LDS.

typedef __attribute__((ext_vector_type(2))) float v2f;
typedef __attribute__((ext_vector_type(8))) float v8f;

#define Bb   32
#define Ll   200
#define Dd   64     // D_IN
#define Hh   64
#define Aa   36
#define NS   0.01f
#define LPAD 208    // 13*16
#define NJT  13     // ceil(200/16)

__device__ __forceinline__ v8f wmma_f32(v2f a, v2f b, v8f c) {
    return __builtin_amdgcn_wmma_f32_16x16x4_f32(false, a, false, b,
                                                 (short)0, c, false, false);
}

__global__ __launch_bounds__(512)
void fused_pair_kernel(const float* __restrict__ x,
                       const float* __restrict__ ln_w,
                       const float* __restrict__ ln_b,
                       const float* __restrict__ w1,
                       const float* __restrict__ b1,
                       const float* __restrict__ w2,
                       const float* __restrict__ b2p,
                       float* __restrict__ out)
{
    __shared__ float hs[LPAD * Dd];          // 53,248 B: h[b] padded to 208 rows

    const int b    = blockIdx.x / NJT;
    const int it   = blockIdx.x % NJT;       // i-tile index
    const int tid  = threadIdx.x;
    const int wave = tid >> 5;
    const int lane = tid & 31;
    const int m    = lane & 15;              // M (A-rows) / N (B,C/D cols) index
    const int half = lane >> 4;              // K sub-pair selector for A/B layouts

    // ---------------- Stage 1: h = x @ ln_w + ln_b into LDS (WMMA) ----------
    // 13 j-tiles x 4 h-col-tiles = 52 16x16 output tiles, striped over 16 waves.
    for (int t = wave; t < NJT * 4; t += 16) {
        const int jt  = t >> 2;
        const int ht  = t & 3;
        const int row = jt * 16 + m;
        const int rowc = (row < Ll) ? row : (Ll - 1);      // clamp; tail masked later
        const float* xr = x + ((size_t)b * Ll + rowc) * Dd;
        const int col = ht * 16 + m;
        v8f acc = {};
#pragma unroll
        for (int ks = 0; ks < 16; ++ks) {
            const int k = ks * 4 + half * 2;
            v2f av, bv;
            av.x = xr[k];
            av.y = xr[k + 1];
            bv.x = ln_w[(k    ) * Hh + col];
            bv.y = ln_w[(k + 1) * Hh + col];
            acc = wmma_f32(av, bv, acc);
        }
        const float lb = ln_b[col];
#pragma unroll
        for (int r = 0; r < 8; ++r) {
            const int rr = jt * 16 + half * 8 + r;         // C/D layout row
            hs[rr * Dd + col] = acc[r] + lb;
        }
    }
    __syncthreads();

    const int i = it * 16 + wave;
    if (i >= Ll) return;                     // wave-uniform exit (tail tile)

    // ---------------- per-lane column constants (a = t*16 + m, padded to 48) --
    const float b2v = b2p[0];
    float w2l[3], vb[3];
#pragma unroll
    for (int t = 0; t < 3; ++t) {
        const int a  = t * 16 + m;
        const bool v = (a < Aa);
        const int ac = v ? a : 0;
        w2l[t] = v ? w2[ac] : 0.f;
        // c_i[a] = h_i . w1b[:,a]   (w1b = rows 64..127 of w1)
        float ci = 0.f;
        for (int k = 0; k < Dd; ++k)
            ci += hs[i * Dd + k] * w1[(Dd + k) * Aa + ac];
        vb[t] = (v ? (ci + b1[ac]) : 0.f);
    }

    // ------- B_i[k,a] = w1a[k,a] + h_i[k]*w1c[k,a], resident in WMMA-B layout
    v2f bm[16][3];
#pragma unroll
    for (int ks = 0; ks < 16; ++ks) {
        const int k0    = ks * 4 + half * 2;
        const float hk0 = hs[i * Dd + k0];
        const float hk1 = hs[i * Dd + k0 + 1];
#pragma unroll
        for (int t = 0; t < 3; ++t) {
            const int a  = t * 16 + m;
            const bool v = (a < Aa);
            const int ac = v ? a : 0;
            const float wa0 = w1[(k0    ) * Aa + ac];
            const float wa1 = w1[(k0 + 1) * Aa + ac];
            const float wc0 = w1[(2 * Dd + k0    ) * Aa + ac];
            const float wc1 = w1[(2 * Dd + k0 + 1) * Aa + ac];
            bm[ks][t].x = v ? (wa0 + hk0 * wc0) : 0.f;
            bm[ks][t].y = v ? (wa1 + hk1 * wc1) : 0.f;
        }
    }

    // ---------------- causal sweep over j-tiles ------------------------------
    float out0 = 0.f, out1 = 0.f;
    for (int jt = 0; jt <= it; ++jt) {
        v8f a0 = {}, a1 = {}, a2 = {};
        const float* hrow = &hs[(jt * 16 + m) * Dd];
#pragma unroll
        for (int ks = 0; ks < 16; ++ks) {
            const int k0 = ks * 4 + half * 2;
            v2f av;
            av.x = hrow[k0];
            av.y = hrow[k0 + 1];
            a0 = wmma_f32(av, bm[ks][0], a0);
            a1 = wmma_f32(av, bm[ks][1], a1);
            a2 = wmma_f32(av, bm[ks][2], a2);
        }
        // bias + leaky-relu + w2 weighting, still in C/D layout
        float s[8];
#pragma unroll
        for (int r = 0; r < 8; ++r) {
            float v0 = a0[r] + vb[0]; v0 = (v0 > 0.f) ? v0 : NS * v0;
            float v1 = a1[r] + vb[1]; v1 = (v1 > 0.f) ? v1 : NS * v1;
            float v2 = a2[r] + vb[2]; v2 = (v2 > 0.f) ? v2 : NS * v2;
            s[r] = v0 * w2l[0] + v1 * w2l[1] + v2 * w2l[2];
        }
        // reduce over the 16 N-lanes of each half (xor 1,2,4,8 stays in-half)
#pragma unroll
        for (int off = 1; off < 16; off <<= 1) {
#pragma unroll
            for (int r = 0; r < 8; ++r) s[r] += __shfl_xor(s[r], off, 32);
        }
        // scores: half0 -> rows jt*16+r, half1 -> rows jt*16+8+r; mask j>i
#pragma unroll
        for (int r = 0; r < 8; ++r) {
            const int j0 = jt * 16 + r;
            const int j1 = j0 + 8;
            float sc0 = __shfl(s[r], 0, 32);
            float sc1 = __shfl(s[r], 16, 32);
            sc0 = (j0 <= i) ? (sc0 + b2v) : 0.f;
            sc1 = (j1 <= i) ? (sc1 + b2v) : 0.f;
            out0 += sc0 * hs[j0 * Dd + lane];
            out1 += sc0 * hs[j0 * Dd + 32 + lane];
            out0 += sc1 * hs[j1 * Dd + lane];
            out1 += sc1 * hs[j1 * Dd + 32 + lane];
        }
    }

    float* op = out + ((size_t)b * Ll + i) * Dd;
    op[lane]      = out0;
    op[lane + 32] = out1;
}

extern "C" void kernel_launch(void* const* d_in, const int* in_sizes, int n_in,
                              void* d_out, int out_size, void* d_ws, size_t ws_size,
                              hipStream_t stream) {
    const float* x    = (const float*)d_in[0];
    const float* ln_w = (const float*)d_in[1];
    const float* ln_b = (const float*)d_in[2];
    const float* w1   = (const float*)d_in[3];
    const float* b1   = (const float*)d_in[4];
    const float* w2   = (const float*)d_in[5];
    const float* b2   = (const float*)d_in[6];
    float* out = (float*)d_out;

    dim3 grid(Bb * NJT);    // 32 batches x 13 i-tiles = 416 workgroups
    dim3 block(512);        // 16 wave32 waves: one wave per output row i
    fused_pair_kernel<<<grid, block, 0, stream>>>(x, ln_w, ln_b, w1, b1, w2, b2, out);
}